// MultiHeadDynamics_63660005261846
// MI455X (gfx1250) — compile-verified
//
#include <hip/hip_runtime.h>

// MultiHeadDynamics on gfx1250: fused low-rank per-head dynamics.
// out = state + DT*((1+cp)*drift - cp*mean_h(drift)),
// drift = (X @ V_h^T) @ U_h^T + X*diag_h + cs*(X - rowmean)^3 + signal.
//
// CDNA5 specifics: V_WMMA_F32_16X16X4_F32 for both GEMMs (fp32 kept end-to-end),
// GLOBAL_LOAD_ASYNC_TO_LDS_B128 (ASYNCcnt) double-buffered X-tile prefetch,
// split WMMA accumulator chains for XDL ILP at 1 workgroup/WGP occupancy.

#define B_TOT 8192
#define D_TOT 4096
#define NH    16
#define DH    256
#define RR    64
#define DT_C  0.05f

typedef float v2f __attribute__((ext_vector_type(2)));
typedef float v8f __attribute__((ext_vector_type(8)));

#define XS_STRIDE 260   // 256 + 4 pad: conflict-free column access (260 % 64 == 4), rows 16B aligned
#define TS_STRIDE 68    // 64 + 4 pad

// LDS layout (float offsets)
#define DS_OFF   0                            // P-store: 16 heads * 16 rows * 256 = 65536 floats
#define XS0_OFF  (DS_OFF + NH * 16 * DH)      // X tile buffer 0: 16 * 260
#define XS1_OFF  (XS0_OFF + 16 * XS_STRIDE)   // X tile buffer 1: 16 * 260
#define TS_OFF   (XS1_OFF + 16 * XS_STRIDE)   // T tile: 16 * 68
#define ACC_OFF  (TS_OFF + 16 * TS_STRIDE)    // drift head-sum: 16 * 256
#define PM_OFF   (ACC_OFF + 16 * DH)          // rowmean partials: 256
#define RM_OFF   (PM_OFF + 256)               // rowmeans: 16
#define SMEM_FLOATS (RM_OFF + 16)             // = 79312 floats = 317248 bytes (< 320KB WGP LDS)

// Issue 4 async B128 loads per thread: X tile (16 rows x 256 f32) of head h -> LDS buffer.
// ISA low-32 bits of a generic LDS pointer are the LDS byte address (flat aperture rule).
__device__ __forceinline__ void async_load_x_tile(const float* __restrict__ state,
                                                  int m0, int h, float* Xbuf, int t) {
    #pragma unroll
    for (int i = 0; i < 4; ++i) {
        int idx = t + i * 256;        // 0..1023 float4s, 64 per row
        int m   = idx >> 6;
        int c4  = idx & 63;
        const float* g = state + (size_t)(m0 + m) * D_TOT + (size_t)h * DH + c4 * 4;
        unsigned lds = (unsigned)(uintptr_t)(&Xbuf[m * XS_STRIDE + c4 * 4]);
        asm volatile("global_load_async_to_lds_b128 %0, %1, off"
                     :: "v"(lds), "v"(g) : "memory");
    }
}

__device__ __forceinline__ void wait_asynccnt0() {
    asm volatile("s_wait_asynccnt 0" ::: "memory");
}

__global__ __launch_bounds__(256, 1)
void MultiHeadDynamics_kernel(const float* __restrict__ state,
                              const float* __restrict__ signal,
                              const float* __restrict__ U,
                              const float* __restrict__ V,
                              const float* __restrict__ diag,
                              const float* __restrict__ cubic_scale_p,
                              const float* __restrict__ coupling_p,
                              float* __restrict__ out)
{
    extern __shared__ float smem[];
    float* Ds  = smem + DS_OFF;
    float* Xs0 = smem + XS0_OFF;
    float* Xs1 = smem + XS1_OFF;
    float* Ts  = smem + TS_OFF;
    float* acc = smem + ACC_OFF;
    float* pm  = smem + PM_OFF;
    float* rm  = smem + RM_OFF;

    const int t    = threadIdx.x;
    const int wave = t >> 5;          // 8 waves of 32 (wave32)
    const int lane = t & 31;
    const int mrow = lane & 15;
    const bool hi  = lane >= 16;
    const int koff = hi ? 2 : 0;      // f32 16x4 A layout: lanes 16-31 hold K+2,K+3
    const int m0   = blockIdx.x * 16; // 16 batch rows per workgroup

    const float cs = *cubic_scale_p;
    const float cp = *coupling_p;
    const float s1 = DT_C * (1.0f + cp);

    // zero the cross-head drift accumulator
    for (int i = t; i < 16 * DH; i += 256) acc[i] = 0.0f;

    // prologue: kick off async X-tile load for head 0
    async_load_x_tile(state, m0, 0, Xs0, t);

    for (int h = 0; h < NH; ++h) {
        float* Xc = (h & 1) ? Xs1 : Xs0;   // current head's tile
        float* Xn = (h & 1) ? Xs0 : Xs1;   // next head's tile (free: last read by head h-1, before a barrier)

        wait_asynccnt0();     // my async loads for head h landed in LDS
        __syncthreads();      // everyone's landed; also fences Ts/acc reuse

        // overlap head h+1's X fetch with both GEMM stages below
        if (h + 1 < NH) async_load_x_tile(state, m0, h + 1, Xn, t);

        // ---- per-row means (for the centered cubic term) ----
        {
            int m = t >> 4, sub = t & 15;
            float p = 0.0f;
            #pragma unroll
            for (int j = 0; j < 16; ++j) p += Xc[m * XS_STRIDE + sub * 16 + j];
            pm[t] = p;
        }
        __syncthreads();
        if (t < 16) {
            float s = 0.0f;
            #pragma unroll
            for (int j = 0; j < 16; ++j) s += pm[t * 16 + j];
            rm[t] = s * (1.0f / DH);
        }
        __syncthreads();

        if (wave < 4) {
            // ---- stage 1 GEMM: T(16x64) = X(16x256) @ V_h^T(256x64) ----
            // two independent accumulator chains (K halves) for XDL ILP
            const int n0 = wave * 16;
            const float* Vh = V + (size_t)h * RR * DH;
            v8f c0 = {}, c1 = {};
            #pragma unroll 2
            for (int k0 = 0; k0 < DH / 2; k0 += 4) {
                v2f a0 = *reinterpret_cast<const v2f*>(&Xc[mrow * XS_STRIDE + k0 + koff]);
                v2f b0 = *reinterpret_cast<const v2f*>(&Vh[(size_t)(n0 + mrow) * DH + k0 + koff]);
                c0 = __builtin_amdgcn_wmma_f32_16x16x4_f32(false, a0, false, b0, (short)0, c0, false, false);
                v2f a1 = *reinterpret_cast<const v2f*>(&Xc[mrow * XS_STRIDE + 128 + k0 + koff]);
                v2f b1 = *reinterpret_cast<const v2f*>(&Vh[(size_t)(n0 + mrow) * DH + 128 + k0 + koff]);
                c1 = __builtin_amdgcn_wmma_f32_16x16x4_f32(false, a1, false, b1, (short)0, c1, false, false);
            }
            const int rbase = hi ? 8 : 0;  // C/D layout: VGPR v = row v (lo lanes) / v+8 (hi lanes)
            #pragma unroll
            for (int v = 0; v < 8; ++v)
                Ts[(rbase + v) * TS_STRIDE + n0 + mrow] = c0[v] + c1[v];
        } else {
            // ---- elementwise phase on waves 4-7 (concurrent with stage 1) ----
            // Ds[h] := x + s1*(x*diag + cs*c^3 + sig);  acc += (x*diag + cs*c^3 + sig)
            int t2 = t - 128;
            const float* dg = diag + (size_t)h * DH;
            float* Dh = Ds + (size_t)h * 16 * DH;
            #pragma unroll 4
            for (int i = 0; i < 32; ++i) {
                int idx = i * 128 + t2;
                int m = idx >> 8;
                int e = idx & 255;
                float x  = Xc[m * XS_STRIDE + e];
                float sg = signal[(size_t)(m0 + m) * D_TOT + (size_t)h * DH + e];
                if (h + 1 < NH)
                    __builtin_prefetch(&signal[(size_t)(m0 + m) * D_TOT + (size_t)(h + 1) * DH + e], 0, 1);
                float cc = x - rm[m];
                float ex = x * dg[e] + cs * cc * cc * cc + sg;
                Dh[m * DH + e]  = x + s1 * ex;
                acc[m * DH + e] += ex;
            }
        }
        __syncthreads();

        // ---- stage 2 GEMM: lin(16x256) = T(16x64) @ U_h^T(64x256) ----
        // each wave computes two 16-wide output tiles; 4 independent WMMA chains
        {
            const float* Uh = U + (size_t)h * DH * RR;
            const int e0 = wave * 32;
            v8f ca0 = {}, ca1 = {}, cb0 = {}, cb1 = {};
            #pragma unroll 2
            for (int k0 = 0; k0 < RR / 2; k0 += 4) {
                v2f a0  = *reinterpret_cast<const v2f*>(&Ts[mrow * TS_STRIDE + k0 + koff]);
                v2f a1  = *reinterpret_cast<const v2f*>(&Ts[mrow * TS_STRIDE + 32 + k0 + koff]);
                v2f ba0 = *reinterpret_cast<const v2f*>(&Uh[(size_t)(e0 + mrow) * RR + k0 + koff]);
                v2f bb0 = *reinterpret_cast<const v2f*>(&Uh[(size_t)(e0 + 16 + mrow) * RR + k0 + koff]);
                v2f ba1 = *reinterpret_cast<const v2f*>(&Uh[(size_t)(e0 + mrow) * RR + 32 + k0 + koff]);
                v2f bb1 = *reinterpret_cast<const v2f*>(&Uh[(size_t)(e0 + 16 + mrow) * RR + 32 + k0 + koff]);
                ca0 = __builtin_amdgcn_wmma_f32_16x16x4_f32(false, a0, false, ba0, (short)0, ca0, false, false);
                cb0 = __builtin_amdgcn_wmma_f32_16x16x4_f32(false, a0, false, bb0, (short)0, cb0, false, false);
                ca1 = __builtin_amdgcn_wmma_f32_16x16x4_f32(false, a1, false, ba1, (short)0, ca1, false, false);
                cb1 = __builtin_amdgcn_wmma_f32_16x16x4_f32(false, a1, false, bb1, (short)0, cb1, false, false);
            }
            const int rbase = hi ? 8 : 0;
            float* Dh = Ds + (size_t)h * 16 * DH;
            #pragma unroll
            for (int v = 0; v < 8; ++v) {
                int r  = rbase + v;
                int ea = e0 + mrow;
                int eb = e0 + 16 + mrow;
                float la = ca0[v] + ca1[v];
                float lb = cb0[v] + cb1[v];
                Dh[r * DH + ea]  += s1 * la;
                acc[r * DH + ea] += la;
                Dh[r * DH + eb]  += s1 * lb;
                acc[r * DH + eb] += lb;
            }
        }
    }
    __syncthreads();

    // ---- final: out = P - DT*cp*mean  (thread t owns column e = t) ----
    const float kf = DT_C * cp * (1.0f / NH);
    for (int m = 0; m < 16; ++m) {
        float mn = acc[m * DH + t] * kf;
        #pragma unroll
        for (int h = 0; h < NH; ++h) {
            out[(size_t)(m0 + m) * D_TOT + (size_t)h * DH + t] =
                Ds[((size_t)h * 16 + m) * DH + t] - mn;
        }
    }
}

extern "C" void kernel_launch(void* const* d_in, const int* in_sizes, int n_in,
                              void* d_out, int out_size, void* d_ws, size_t ws_size,
                              hipStream_t stream) {
    (void)in_sizes; (void)n_in; (void)out_size; (void)d_ws; (void)ws_size;
    const float* state  = (const float*)d_in[0];
    const float* signal = (const float*)d_in[1];
    const float* U      = (const float*)d_in[2];
    const float* V      = (const float*)d_in[3];
    const float* diag   = (const float*)d_in[4];
    const float* cs     = (const float*)d_in[5];
    const float* cp     = (const float*)d_in[6];
    float* out = (float*)d_out;

    const size_t shmem = (size_t)SMEM_FLOATS * sizeof(float); // 317248 B
    hipFuncSetAttribute((const void*)MultiHeadDynamics_kernel,
                        hipFuncAttributeMaxDynamicSharedMemorySize, (int)shmem);
    MultiHeadDynamics_kernel<<<B_TOT / 16, 256, shmem, stream>>>(
        state, signal, U, V, diag, cs, cp, out);
}